// Model_85444079386771
// MI455X (gfx1250) — compile-verified
//
#include <hip/hip_runtime.h>
#include <hip/hip_bf16.h>

// ---------------------------------------------------------------------------
// GAT-style 4-layer GNN for MI455X (gfx1250), wave32 + WMMA f32 16x16x4.
// ---------------------------------------------------------------------------

typedef float v2f __attribute__((ext_vector_type(2)));
typedef float v8f __attribute__((ext_vector_type(8)));

#define NN   100000
#define NE   1600000
#define IND  256
#define HIDD 128
#define OUTD 32
#define NEG_SLOPE 0.2f

// ---------------------------------------------------------------------------
// Generic row-strip GEMM: D[M x NC] = A[M x K] @ B[K x NC] using
// V_WMMA_F32_16X16X4_F32.  One wave computes a 16-row strip, full NC width
// (split in <=128-col passes to bound accumulator VGPRs at 64).
//
// B is staged fully into LDS once per block (max 128 KB < 320 KB WGP LDS) in
// a PAIR-INTERLEAVED layout: element B[k][n] lives at
//     smem[((k>>1)*NC + n)*2 + (k&1)]
// so the two values a WMMA B-fragment needs (rows k,k+1, same column) are one
// aligned 8-byte word -> ds_load_b64/ds_load_2addr_b64 into even VGPR pairs.
//
// The k-loop is software-pipelined with TWO fragment buffers (k unrolled by
// 8): loads for k-step i+1 are issued before the WMMA burst of step i, so
// their lifetimes overlap, the register allocator cannot recycle one quad,
// and the LDS latency hides behind the previous WMMA burst instead of a
// s_wait_dscnt 0x0 stall per fragment pair.
//
// ELU optionally applied to A elements on load (fuses the reference's elu).
// ---------------------------------------------------------------------------
template <int K, int NC, bool BT, bool ELU>
__global__ __launch_bounds__(128) void gemm_wmma(const float* __restrict__ A,
                                                 const float* __restrict__ Bm,
                                                 float* __restrict__ D, int M) {
  extern __shared__ float smem[];  // K*NC floats, pair-interleaved

  const int tid = threadIdx.x;
  for (int i = tid; i < K * NC; i += blockDim.x) {
    const int k = i / NC;
    const int n = i - k * NC;
    smem[(((k >> 1) * NC + n) << 1) | (k & 1)] =
        BT ? Bm[(size_t)n * K + k] : Bm[i];
  }
  __syncthreads();

  const int lane  = tid & 31;
  const int wv    = tid >> 5;
  const int strip = blockIdx.x * (blockDim.x >> 5) + wv;
  const int row0  = strip * 16;
  if (row0 >= M) return;  // M is a multiple of 16: strips are all-or-nothing

  const int m    = lane & 15;          // output row within strip / B column
  const int koff = (lane >> 4) << 1;   // 0 or 2 : A/B k-pair per half-wave
  const float* arow = A + (size_t)(row0 + m) * K + koff;  // 8B aligned

  constexpr int NH    = (NC > 128) ? 128 : NC;  // columns per pass
  constexpr int NPASS = NC / NH;
  constexpr int NT    = NH / 16;                // 16-col WMMA tiles per pass
  static_assert(K % 8 == 0, "k-loop unrolled by 8");

  for (int np = 0; np < NPASS; ++np) {
    v8f acc[NT];
#pragma unroll
    for (int t = 0; t < NT; ++t)
#pragma unroll
      for (int i = 0; i < 8; ++i) acc[t][i] = 0.0f;

    // B fragment loader: rows (kk+koff, kk+koff+1), columns np*NH + m + 16t
    auto loadB = [&](int kk, v2f* dst) {
      const float* bp = smem + ((((kk + koff) >> 1) * NC + np * NH + m) << 1);
#pragma unroll
      for (int t = 0; t < NT; ++t)
        dst[t] = *reinterpret_cast<const v2f*>(bp + t * 32);  // ds_load_b64
    };
    // A fragment load (+optional fused ELU) and WMMA burst over one k-step.
    auto mmaStep = [&](int kk, const v2f* bv) {
      v2f av = *reinterpret_cast<const v2f*>(arow + kk);  // global_load_b64
      if (ELU) {
        av[0] = av[0] > 0.0f ? av[0] : (expf(av[0]) - 1.0f);
        av[1] = av[1] > 0.0f ? av[1] : (expf(av[1]) - 1.0f);
      }
#pragma unroll
      for (int t = 0; t < NT; ++t)
        acc[t] = __builtin_amdgcn_wmma_f32_16x16x4_f32(
            false, av, false, bv[t], (short)0, acc[t], false, false);
    };

    v2f bv0[NT], bv1[NT];
    loadB(0, bv0);
    for (int k0 = 0; k0 < K; k0 += 8) {
      loadB(k0 + 4, bv1);            // prefetch next step's fragments
      mmaStep(k0, bv0);              // consume current (latency hidden)
      if (k0 + 8 < K) loadB(k0 + 8, bv0);
      mmaStep(k0 + 4, bv1);
    }

    const int mbase = (lane >> 4) * 8;  // D layout: lanes 16-31 hold M=8..15
#pragma unroll
    for (int t = 0; t < NT; ++t) {
#pragma unroll
      for (int v = 0; v < 8; ++v) {
        D[(size_t)(row0 + mbase + v) * NC + np * NH + t * 16 + m] = acc[t][v];
      }
    }
  }
}

// ---------------------------------------------------------------------------
// Per-node attention logits: as1[i] = xs1[i,:].a_src ; ad1[i] = xs1[i,:].a_dst
// One wave32 per node; each lane owns 4 contiguous channels (float4 loads).
// ---------------------------------------------------------------------------
__global__ void node_dots(const float* __restrict__ xs1,
                          const float* __restrict__ a_s,
                          const float* __restrict__ a_d,
                          float* __restrict__ as1, float* __restrict__ ad1,
                          int n) {
  const int gw   = (blockIdx.x * blockDim.x + threadIdx.x) >> 5;
  const int lane = threadIdx.x & 31;
  if (gw >= n) return;
  const float4 v  = ((const float4*)(xs1 + (size_t)gw * HIDD))[lane];
  const float4 cs = ((const float4*)a_s)[lane];
  const float4 cd = ((const float4*)a_d)[lane];
  float s = v.x * cs.x + v.y * cs.y + v.z * cs.z + v.w * cs.w;
  float d = v.x * cd.x + v.y * cd.y + v.z * cd.z + v.w * cd.w;
#pragma unroll
  for (int off = 16; off > 0; off >>= 1) {
    s += __shfl_down(s, off, 32);
    d += __shfl_down(d, off, 32);
  }
  if (lane == 0) {
    as1[gw] = s;
    ad1[gw] = d;
  }
}

// Monotone float<->uint encoding for atomic segment-max.
__device__ __forceinline__ unsigned f2key(float f) {
  const unsigned b = __float_as_uint(f);
  return (b & 0x80000000u) ? ~b : (b | 0x80000000u);
}
__device__ __forceinline__ float key2f(unsigned u) {
  return __uint_as_float((u & 0x80000000u) ? (u & 0x7FFFFFFFu) : ~u);
}

// e = leaky_relu(as1[src]+ad1[dst]); segment max via encoded atomicMax.
__global__ void edge_logits(const int* __restrict__ ei,
                            const float* __restrict__ as1,
                            const float* __restrict__ ad1,
                            float* __restrict__ ebuf,
                            unsigned* __restrict__ mbuf, int E) {
  const int e = blockIdx.x * blockDim.x + threadIdx.x;
  if (e >= E) return;
  const int s = ei[e];
  const int d = ei[E + e];
  float v = as1[s] + ad1[d];
  v = v > 0.0f ? v : NEG_SLOPE * v;
  ebuf[e] = v;
  atomicMax(&mbuf[d], f2key(v));
}

// w = exp(e - m[dst]); segment sum via atomicAdd.  ebuf rewritten in place.
__global__ void edge_expw(const int* __restrict__ ei,
                          const unsigned* __restrict__ mbuf,
                          float* __restrict__ ebuf, float* __restrict__ sbuf,
                          int E) {
  const int e = blockIdx.x * blockDim.x + threadIdx.x;
  if (e >= E) return;
  const int d = ei[E + e];
  const float w = expf(ebuf[e] - key2f(mbuf[d]));
  ebuf[e] = w;
  atomicAdd(&sbuf[d], w);
}

// Weighted scatter-add of 128-f32 feature rows; one wave32 per edge.
// Lane owns 4 contiguous channels: gather is one coalesced global_load_b128
// per lane (512 B per edge), scatter is 4 global_atomic_add_f32.
// FIRST: compute alpha = w / s[dst] and cache it; else reuse cached alpha
// (the reference ties conv3's attention to conv1's).
template <bool FIRST>
__global__ void edge_aggregate(const int* __restrict__ ei,
                               const float* __restrict__ feat,
                               const float* __restrict__ wb,
                               const float* __restrict__ sb,
                               float* __restrict__ alpha,
                               float* __restrict__ agg, int E) {
  const int gw   = (blockIdx.x * blockDim.x + threadIdx.x) >> 5;
  const int lane = threadIdx.x & 31;
  if (gw >= E) return;
  const int s = ei[gw];
  const int d = ei[E + gw];
  float a;
  if (FIRST) {
    a = wb[gw] / sb[d];
    if (lane == 0) alpha[gw] = a;
  } else {
    a = alpha[gw];
  }
  const float4 v = ((const float4*)(feat + (size_t)s * HIDD))[lane];
  float* ar = agg + (size_t)d * HIDD + lane * 4;
  atomicAdd(ar + 0, v.x * a);
  atomicAdd(ar + 1, v.y * a);
  atomicAdd(ar + 2, v.z * a);
  atomicAdd(ar + 3, v.w * a);
}

// ---------------------------------------------------------------------------
extern "C" void kernel_launch(void* const* d_in, const int* in_sizes, int n_in,
                              void* d_out, int out_size, void* d_ws,
                              size_t ws_size, hipStream_t stream) {
  const float* x   = (const float*)d_in[0];
  const float* W1  = (const float*)d_in[1];
  const float* a1s = (const float*)d_in[2];
  const float* a1d = (const float*)d_in[3];
  const float* W2  = (const float*)d_in[4];
  const int*   ei  = (const int*)d_in[5];  // [2, NE] flat (src row, dst row)

  float* out = (float*)d_out;
  float* h2  = out;                     // [NN, 32]
  float* h4  = out + (size_t)NN * OUTD; // [NN, 256]

  // Workspace layout (floats)
  float* ws    = (float*)d_ws;
  float* xs1   = ws;                         // NN*128 (reused as xs3)
  float* agg   = xs1 + (size_t)NN * HIDD;    // NN*128 (agg1, reused as agg3)
  float* as1   = agg + (size_t)NN * HIDD;    // NN
  float* ad1   = as1 + NN;                   // NN
  float* sbuf  = ad1 + NN;                   // NN
  unsigned* mbuf = (unsigned*)(sbuf + NN);   // NN
  float* ebuf  = (float*)(mbuf + NN);        // NE (e -> w in place)
  float* alpha = ebuf + NE;                  // NE

  // Zero the accumulators / segment buffers (sbuf & mbuf are adjacent).
  hipMemsetAsync(agg, 0, (size_t)NN * HIDD * sizeof(float), stream);
  hipMemsetAsync(sbuf, 0, (size_t)2 * NN * sizeof(float), stream);

  const int strips  = NN / 16;             // 6250 (NN % 16 == 0)
  const int gblocks = (strips + 3) / 4;    // 4 waves per 128-thread block
  const int eblocks = (NE + 255) / 256;    // thread-per-edge kernels
  const int ewarpsB = NE / 8;              // wave-per-edge kernels (256 thr)
  const int nwarpsB = (NN * 32 + 255) / 256;

  // conv1 linear: xs1 = x @ W1    (K=256, NC=128, LDS = 128 KB)
  gemm_wmma<IND, HIDD, false, false>
      <<<gblocks, 128, IND * HIDD * sizeof(float), stream>>>(x, W1, xs1, NN);

  // attention logits
  node_dots<<<nwarpsB, 256, 0, stream>>>(xs1, a1s, a1d, as1, ad1, NN);

  // segment softmax over dst
  edge_logits<<<eblocks, 256, 0, stream>>>(ei, as1, ad1, ebuf, mbuf, NE);
  edge_expw<<<eblocks, 256, 0, stream>>>(ei, mbuf, ebuf, sbuf, NE);

  // conv1 aggregation (caches alpha for conv3)
  edge_aggregate<true>
      <<<ewarpsB, 256, 0, stream>>>(ei, xs1, ebuf, sbuf, alpha, agg, NE);

  // conv2: h2 = elu(agg1) @ W2   (K=128, NC=32, LDS = 16 KB)
  gemm_wmma<HIDD, OUTD, false, true>
      <<<gblocks, 128, HIDD * OUTD * sizeof(float), stream>>>(agg, W2, h2, NN);

  // conv3 linear: xs3 = h2 @ W2^T  (B staged transposed; xs3 reuses xs1)
  gemm_wmma<OUTD, HIDD, true, false>
      <<<gblocks, 128, OUTD * HIDD * sizeof(float), stream>>>(h2, W2, xs1, NN);

  // conv3 aggregation with tied attention
  hipMemsetAsync(agg, 0, (size_t)NN * HIDD * sizeof(float), stream);
  edge_aggregate<false>
      <<<ewarpsB, 256, 0, stream>>>(ei, xs1, ebuf, sbuf, alpha, agg, NE);

  // conv4: h4 = elu(agg3) @ W1^T  (K=128, NC=256, LDS = 128 KB, 2 col passes)
  gemm_wmma<HIDD, IND, true, true>
      <<<gblocks, 128, HIDD * IND * sizeof(float), stream>>>(agg, W1, h4, NN);
}